// ROIAlign_81174881894441
// MI455X (gfx1250) — compile-verified
//
#include <hip/hip_runtime.h>

// CDNA5 / gfx1250 ROI-Align (FPN, 4 levels, OUT=7, C=256) with the
// x-interpolation expressed as a K=16 GEMM via V_WMMA_F32_16X16X4_F32.

typedef float v2f __attribute__((ext_vector_type(2)));
typedef float v8f __attribute__((ext_vector_type(8)));

#define CCH 256

// Reference-exact per-axis bilinear terms (handles the clip semantics where
// out-of-range samples get zero weight because x1f==xc==x0f after clipping).
__device__ __forceinline__ void axis_interp(float v, int limit, int& lo, int& hi,
                                            float& wlo, float& whi) {
  int i0 = (int)floorf(v);
  int i1 = i0 + 1;
  float vc = fminf(fmaxf(v, 0.0f), (float)(limit - 1));
  lo = min(max(i0, 0), limit - 1);
  hi = min(max(i1, 0), limit - 1);
  wlo = (float)hi - vc;   // (x1f - xc)
  whi = vc - (float)lo;   // (xc - x0f)
}

__global__ __launch_bounds__(256)
void roi_align_wmma_kernel(const float* __restrict__ f0, const float* __restrict__ f1,
                           const float* __restrict__ f2, const float* __restrict__ f3,
                           const float* __restrict__ proposals,
                           float* __restrict__ out, int nPerBatch) {
  // Per-ROI metadata in LDS (computed once per block).
  __shared__ int   sXcol[16];               // 14 column slots (+2 padded, clamped)
  __shared__ float sWxLo[8], sWxHi[8];      // x weights per output column
  __shared__ int   sYlo[8], sYhi[8];        // y source rows per output row
  __shared__ float sWyLo[8], sWyHi[8];      // y weights per output row
  __shared__ int   sLvl, sB;

  const int r   = blockIdx.x;
  const int tid = threadIdx.x;

  if (tid < 8) {
    const float x0 = proposals[r * 4 + 0];
    const float y0 = proposals[r * 4 + 1];
    const float x1 = proposals[r * 4 + 2];
    const float y1 = proposals[r * 4 + 3];
    const float pw = x1 - x0, ph = y1 - y0;
    int lvl = (int)floorf(2.0f + log2f(sqrtf(pw * ph) * (1.0f / 224.0f)));
    lvl = min(max(lvl, 0), 3);
    const int   W     = 256 >> lvl;
    const float scale = 1.0f / (float)(4 << lvl);   // 0.25 / 2^lvl
    const float bx0 = x0 * scale, by0 = y0 * scale;
    const float bx1 = x1 * scale, by1 = y1 * scale;
    const float wu = (bx1 - bx0) * (1.0f / 7.0f);
    const float hu = (by1 - by0) * (1.0f / 7.0f);
    if (tid < 7) {
      const int g = tid;
      int lo, hi; float wlo, whi;
      axis_interp(((float)g + 0.5f) * wu + bx0, W, lo, hi, wlo, whi);
      sXcol[2 * g] = lo; sXcol[2 * g + 1] = hi;
      sWxLo[g] = wlo;    sWxHi[g] = whi;
      axis_interp(((float)g + 0.5f) * hu + by0, W, lo, hi, wlo, whi);
      sYlo[g] = lo; sYhi[g] = hi;
      sWyLo[g] = wlo; sWyHi[g] = whi;
    } else {
      sXcol[14] = 0; sXcol[15] = 0;  // padded slots: valid address, weight 0
      sLvl = lvl;
      sB   = r / nPerBatch;
    }
  }
  __syncthreads();

  const int lvl = sLvl;
  const int b   = sB;
  const int W   = 256 >> lvl;
  const int HW  = W * W;
  const float* feat = (lvl == 0) ? f0 : (lvl == 1) ? f1 : (lvl == 2) ? f2 : f3;

  const int ln     = tid & 31;
  const int wave   = tid >> 5;
  const int cLane  = ln & 15;          // channel within 16-block == WMMA N, and A/B M
  const int khalf  = (ln >> 4) * 2;    // K-half offset: lanes 0-15 -> K0,1; 16-31 -> K2,3

  // Per-lane, per-ROI constants: 8 gather column offsets + 8 A-matrix weights.
  int   colOff[8];
  float aW[8];
#pragma unroll
  for (int kc = 0; kc < 4; ++kc) {
#pragma unroll
    for (int q = 0; q < 2; ++q) {
      const int s = 4 * kc + khalf + q;          // column slot for this K element
      colOff[2 * kc + q] = sXcol[s];
      float w = 0.0f;
      if (cLane < 7) {                           // A[m=ox, s]: 2 nonzeros per row
        if (s == 2 * cLane)          w = sWxLo[cLane];
        else if (s == 2 * cLane + 1) w = sWxHi[cLane];
      }
      aW[2 * kc + q] = w;
    }
  }

  const float* planeBase = feat + (size_t)(b * CCH + cLane) * (size_t)HW;

  // 7 (oy) x 16 (channel blocks) = 112 tiles; 8 waves -> 14 tiles each (uniform,
  // EXEC all-ones at every WMMA as the ISA requires).
  for (int t = wave; t < 7 * 16; t += 8) {
    const int oy = t >> 4;
    const int cb = t & 15;
    const int   yLo = sYlo[oy], yHi = sYhi[oy];
    const float wyl = sWyLo[oy], wyh = sWyHi[oy];

    const float* plane = planeBase + (size_t)(cb * 16) * (size_t)HW;
    const float* rowL  = plane + yLo * W;
    const float* rowH  = plane + yHi * W;

    float vLo[8], vHi[8];
#pragma unroll
    for (int j = 0; j < 8; ++j) {      // 16 gathers, clustered for MLP
      vLo[j] = rowL[colOff[j]];
      vHi[j] = rowH[colOff[j]];
    }
    float vy[8];
#pragma unroll
    for (int j = 0; j < 8; ++j)        // y-interp on VALU
      vy[j] = fmaf(wyh, vHi[j], wyl * vLo[j]);

    // x-interp as GEMM: D[ox, c] += A[ox, k] * VY[k, c], K=16 via 4 chained WMMAs.
    v8f d = {0.f, 0.f, 0.f, 0.f, 0.f, 0.f, 0.f, 0.f};
#pragma unroll
    for (int kc = 0; kc < 4; ++kc) {
      v2f a;  a.x  = aW[2 * kc]; a.y  = aW[2 * kc + 1];
      v2f bb; bb.x = vy[2 * kc]; bb.y = vy[2 * kc + 1];
      d = __builtin_amdgcn_wmma_f32_16x16x4_f32(false, a, false, bb,
                                                (short)0, d, false, false);
    }

    // D rows 0..6 (ox) live in vgprs 0..6 of lanes 0-15. Non-temporal stores so
    // the 51MB output stream does not evict the ~170MB feature set from L2.
    if ((ln >> 4) == 0) {
      const size_t o = ((size_t)(r * CCH + cb * 16 + cLane) * 7 + oy) * 7;
#pragma unroll
      for (int v = 0; v < 7; ++v)
        __builtin_nontemporal_store(d[v], out + o + v);
    }
  }
}

extern "C" void kernel_launch(void* const* d_in, const int* in_sizes, int n_in,
                              void* d_out, int out_size, void* d_ws, size_t ws_size,
                              hipStream_t stream) {
  (void)n_in; (void)out_size; (void)d_ws; (void)ws_size;
  const float* f0 = (const float*)d_in[0];
  const float* f1 = (const float*)d_in[1];
  const float* f2 = (const float*)d_in[2];
  const float* f3 = (const float*)d_in[3];
  const float* proposals = (const float*)d_in[4];
  float* out = (float*)d_out;

  // feat0 is (B, 256, 256, 256) -> recover B; proposals is (B, N, 4) -> N.
  long planesBC = (long)in_sizes[0] / (256L * 256L);   // B*C
  int B = (int)(planesBC / 256L);
  if (B < 1) B = 1;
  int N = in_sizes[4] / (4 * B);
  int R = B * N;

  roi_align_wmma_kernel<<<dim3(R), dim3(256), 0, stream>>>(
      f0, f1, f2, f3, proposals, out, N);
}